// CrossImageConsistency_69767448756721
// MI455X (gfx1250) — compile-verified
//
#include <hip/hip_runtime.h>
#include <hip/hip_bf16.h>

// Problem constants (from reference): B=16, C=256, H=W=64 -> S=4096, NH=8, HD=32
#define BB 16
#define SS 4096
#define CC 256

typedef __attribute__((ext_vector_type(2))) float v2f;
typedef __attribute__((ext_vector_type(8))) float v8f;

// ---------------------------------------------------------------------------
// Kernel 1: fused Q/K/V GEMM.  y = xf @ W^T + b  for W in {Wq,Wk,Wv}.
// Block = 32 consecutive positions (one b), all 256 outputs, all 3 matrices.
// A tile (xf 32x256) staged in LDS; f32 WMMA 16x16x4, K=256 -> 64 steps.
// Each B fragment (weights) is reused by 2 WMMAs (two M-tiles) -> halves
// weight streaming from L2 vs a 16-row tile.
// ---------------------------------------------------------------------------
__global__ __launch_bounds__(256) void qkv_gemm_kernel(
    const float* __restrict__ x,
    const float* __restrict__ Wq, const float* __restrict__ bq,
    const float* __restrict__ Wk, const float* __restrict__ bk,
    const float* __restrict__ Wv, const float* __restrict__ bv,
    float* __restrict__ q, float* __restrict__ k, float* __restrict__ v)
{
    __shared__ float A[CC * 33];               // A[c*33 + m], m in [0,32)
    const int g  = blockIdx.x;                 // 0..2047
    const int b  = g >> 7;                     // 128 tiles per image
    const int s0 = (g & 127) * 32;
    const int t  = threadIdx.x;

    // Load xf tile: xf[s0+m, c] = x[b, c, s0+m]; 32 lanes read 128B contiguous.
    {
        const int m  = t & 31;
        const int cb = t >> 5;                 // 0..7
        const float* xb = x + ((size_t)b * CC) * SS + s0 + m;
        #pragma unroll
        for (int it = 0; it < 32; ++it) {
            const int c = cb + it * 8;
            A[c * 33 + m] = xb[(size_t)c * SS];
        }
    }
    __syncthreads();

    const int lane = t & 31;
    const int wave = t >> 5;                   // 0..7
    const int half = lane >> 4;                // K half-offset select
    const int nlo  = lane & 15;                // N (and M for A-frag) lane index
    const int nb0  = wave * 2;                 // this wave's two N-tiles

    const float* Wm[3]   = {Wq, Wk, Wv};
    const float* bias[3] = {bq, bk, bv};
    float*       outp[3] = {q, k, v};

    v8f zero = {};
    v8f acc[3][2][2];                          // [matrix][n-tile][m-tile]
    #pragma unroll
    for (int mtx = 0; mtx < 3; ++mtx)
        #pragma unroll
        for (int j = 0; j < 2; ++j) {
            acc[mtx][j][0] = zero;
            acc[mtx][j][1] = zero;
        }

    #pragma unroll 2
    for (int kb = 0; kb < 64; ++kb) {
        const int kk = kb * 4 + half * 2;      // this lane's K pair
        v2f a0, a1;                            // A frags for m-tiles 0 and 1
        a0.x = A[kk * 33 + nlo];
        a0.y = A[(kk + 1) * 33 + nlo];
        a1.x = A[kk * 33 + 16 + nlo];
        a1.y = A[(kk + 1) * 33 + 16 + nlo];
        #pragma unroll
        for (int mtx = 0; mtx < 3; ++mtx) {
            #pragma unroll
            for (int j = 0; j < 2; ++j) {
                const int row = (nb0 + j) * 16 + nlo;   // W row = output n
                const float2 bw = *(const float2*)(Wm[mtx] + (size_t)row * CC + kk);
                v2f bf; bf.x = bw.x; bf.y = bw.y;       // B[k,n] = W[n,k]
                acc[mtx][j][0] = __builtin_amdgcn_wmma_f32_16x16x4_f32(
                    false, a0, false, bf, (short)0, acc[mtx][j][0], false, false);
                acc[mtx][j][1] = __builtin_amdgcn_wmma_f32_16x16x4_f32(
                    false, a1, false, bf, (short)0, acc[mtx][j][1], false, false);
            }
        }
    }

    // Epilogue: + bias, store to [B*S, 256] buffers (coalesced 64B per half-wave)
    #pragma unroll
    for (int mtx = 0; mtx < 3; ++mtx) {
        #pragma unroll
        for (int j = 0; j < 2; ++j) {
            const int n = (nb0 + j) * 16 + nlo;
            const float bv_ = bias[mtx][n];
            float* op = outp[mtx] + ((size_t)b * SS + s0) * CC + n;
            #pragma unroll
            for (int mt = 0; mt < 2; ++mt) {
                #pragma unroll
                for (int r = 0; r < 8; ++r) {
                    const int m = mt * 16 + r + half * 8;  // C/D: M = r + 8*half
                    op[(size_t)m * CC] = acc[mtx][j][mt][r] + bv_;
                }
            }
        }
    }
}

// ---------------------------------------------------------------------------
// Kernel 2: ksum[s,c] = sum_b k[b,s,c]   (deterministic, no atomics)
// ---------------------------------------------------------------------------
__global__ __launch_bounds__(256) void ksum_kernel(
    const float* __restrict__ k, float* __restrict__ ksum)
{
    const int idx = blockIdx.x * 256 + threadIdx.x;  // 0 .. S*C-1
    float s = 0.0f;
    #pragma unroll
    for (int b = 0; b < BB; ++b) s += k[(size_t)b * (SS * CC) + idx];
    ksum[idx] = s;
}

// ---------------------------------------------------------------------------
// Kernel 3: per-position 8x8 attention + softmax + scrambled store.
// One wave per (b,s). Writes directly into [B,H,W,C] "torch-view" layout:
//   (s,i,d) -> h=(s>>3)&63, w=(s&7)*8+i, c=(s>>9)*32+d
// ---------------------------------------------------------------------------
__global__ __launch_bounds__(256) void attn_kernel(
    const float* __restrict__ q, const float* __restrict__ ksum,
    const float* __restrict__ v, float* __restrict__ y)
{
    __shared__ float sh[8][3 * 264 + 64];      // per wave: q,ks,v (stride 33) + p[64]
    const int t = threadIdx.x, wave = t >> 5, lane = t & 31;
    const int p = blockIdx.x * 8 + wave;       // 0..65535
    const int b = p >> 12, s = p & 4095;

    float* qs = sh[wave];
    float* ks = qs + 264;
    float* vs = qs + 528;
    float* ps = qs + 792;

    const size_t qoff = (size_t)p * CC;
    const size_t koff = (size_t)s * CC;
    #pragma unroll
    for (int i = 0; i < 8; ++i) {
        qs[i * 33 + lane] = q[qoff + i * 32 + lane];
        ks[i * 33 + lane] = ksum[koff + i * 32 + lane];
        vs[i * 33 + lane] = v[qoff + i * 32 + lane];
    }
    __syncthreads();

    // Each lane owns 2 logits: i = lane>>2 ; j = (lane&3)*2 + {0,1}
    const int i  = lane >> 2;
    const int j0 = (lane & 3) * 2;
    float d0 = 0.0f, d1 = 0.0f;
    #pragma unroll
    for (int d = 0; d < 32; ++d) {
        const float qv = qs[i * 33 + d];
        d0 += qv * ks[j0 * 33 + d];
        d1 += qv * ks[(j0 + 1) * 33 + d];
    }
    // reference: softmax(attn / SCALE), SCALE = HD^-0.5 -> multiply by sqrt(32)
    const float INV_SCALE = 5.65685424949238019521f;
    d0 *= INV_SCALE; d1 *= INV_SCALE;

    float mx = fmaxf(d0, d1);
    mx = fmaxf(mx, __shfl_xor(mx, 1));
    mx = fmaxf(mx, __shfl_xor(mx, 2));
    const float e0 = expf(d0 - mx), e1 = expf(d1 - mx);
    float sum = e0 + e1;
    sum += __shfl_xor(sum, 1);
    sum += __shfl_xor(sum, 2);
    const float inv = 1.0f / sum;
    ps[i * 8 + j0]     = e0 * inv;
    ps[i * 8 + j0 + 1] = e1 * inv;
    __syncthreads();

    // out[i][d], d = lane; scrambled coalesced store (32 contiguous floats/row)
    const int n = s >> 9, h = (s >> 3) & 63, whi = s & 7;
    float* yb = y + (((size_t)b * 64 + h) * 64) * CC + (size_t)n * 32 + lane;
    #pragma unroll
    for (int ii = 0; ii < 8; ++ii) {
        float o = 0.0f;
        #pragma unroll
        for (int jj = 0; jj < 8; ++jj) o += ps[ii * 8 + jj] * vs[jj * 33 + lane];
        const int w = whi * 8 + ii;
        yb[(size_t)w * CC] = o;
    }
}

// ---------------------------------------------------------------------------
// Kernel 4: z = y @ Wo^T + bo ; LayerNorm over C ; out = transpose + x
// M=32 tile: each B fragment reused by 2 WMMAs; LN fused via LDS z-tile;
// NCHW store coalesced (lanes map to 32 consecutive s -> 128B).
// ---------------------------------------------------------------------------
__global__ __launch_bounds__(256) void proj_ln_kernel(
    const float* __restrict__ y,
    const float* __restrict__ Wo, const float* __restrict__ bo,
    const float* __restrict__ ln_g, const float* __restrict__ ln_b,
    const float* __restrict__ x, float* __restrict__ out)
{
    __shared__ float A[CC * 33];               // A[c*33 + m], m in [0,32)
    __shared__ float Z[32 * 257];
    __shared__ float mean_s[32], rstd_s[32];

    const int r0 = blockIdx.x * 32;            // global row = b*4096 + s
    const int t  = threadIdx.x;

    // Load y tile [32][256] (rows contiguous) -> A[c*33 + m]
    {
        const int m0 = t >> 4;                 // 0..15
        const int cb = t & 15;
        #pragma unroll
        for (int mh = 0; mh < 2; ++mh) {
            const int m = m0 + mh * 16;
            const float* yr = y + (size_t)(r0 + m) * CC;
            #pragma unroll
            for (int it = 0; it < 16; ++it) {
                const int c = cb + it * 16;
                A[c * 33 + m] = yr[c];
            }
        }
    }
    __syncthreads();

    const int lane = t & 31, wave = t >> 5;
    const int half = lane >> 4, nlo = lane & 15;
    const int nb0  = wave * 2;

    v8f zero = {};
    v8f acc[2][2];                             // [n-tile][m-tile]
    acc[0][0] = zero; acc[0][1] = zero;
    acc[1][0] = zero; acc[1][1] = zero;

    #pragma unroll 2
    for (int kb = 0; kb < 64; ++kb) {
        const int kk = kb * 4 + half * 2;
        v2f a0, a1;
        a0.x = A[kk * 33 + nlo];
        a0.y = A[(kk + 1) * 33 + nlo];
        a1.x = A[kk * 33 + 16 + nlo];
        a1.y = A[(kk + 1) * 33 + 16 + nlo];
        #pragma unroll
        for (int j = 0; j < 2; ++j) {
            const int row = (nb0 + j) * 16 + nlo;
            const float2 bw = *(const float2*)(Wo + (size_t)row * CC + kk);
            v2f bf; bf.x = bw.x; bf.y = bw.y;
            acc[j][0] = __builtin_amdgcn_wmma_f32_16x16x4_f32(
                false, a0, false, bf, (short)0, acc[j][0], false, false);
            acc[j][1] = __builtin_amdgcn_wmma_f32_16x16x4_f32(
                false, a1, false, bf, (short)0, acc[j][1], false, false);
        }
    }

    // z + bias into LDS (padded stride 257)
    #pragma unroll
    for (int j = 0; j < 2; ++j) {
        const int n = (nb0 + j) * 16 + nlo;
        const float bv_ = bo[n];
        #pragma unroll
        for (int mt = 0; mt < 2; ++mt) {
            #pragma unroll
            for (int r = 0; r < 8; ++r)
                Z[(mt * 16 + r + half * 8) * 257 + n] = acc[j][mt][r] + bv_;
        }
    }
    __syncthreads();

    // Row stats: 8 threads per row (32 rows), shuffle-reduce over 8-lane groups
    {
        const int m = t >> 3, part = t & 7;
        float s1 = 0.0f, s2 = 0.0f;
        #pragma unroll
        for (int cc = 0; cc < 32; ++cc) {
            const float z = Z[m * 257 + cc * 8 + part];
            s1 += z; s2 += z * z;
        }
        s1 += __shfl_xor(s1, 1); s2 += __shfl_xor(s2, 1);
        s1 += __shfl_xor(s1, 2); s2 += __shfl_xor(s2, 2);
        s1 += __shfl_xor(s1, 4); s2 += __shfl_xor(s2, 4);
        if (part == 0) {
            const float mean = s1 * (1.0f / 256.0f);
            const float var  = s2 * (1.0f / 256.0f) - mean * mean;
            mean_s[m] = mean;
            rstd_s[m] = rsqrtf(var + 1e-6f);
        }
    }
    __syncthreads();

    // Normalize + residual + NHWC -> NCHW store (lanes = 32 consecutive s)
    {
        const int m  = t & 31;
        const int cb = t >> 5;                 // 0..7
        const int b  = r0 >> 12;
        const int s  = (r0 & 4095) + m;
        const float mean = mean_s[m], rstd = rstd_s[m];
        #pragma unroll
        for (int it = 0; it < 32; ++it) {
            const int c = cb + it * 8;
            const size_t oidx = ((size_t)b * CC + c) * SS + s;
            const float zz = (Z[m * 257 + c] - mean) * rstd * ln_g[c] + ln_b[c];
            out[oidx] = zz + x[oidx];
        }
    }
}

// ---------------------------------------------------------------------------
extern "C" void kernel_launch(void* const* d_in, const int* in_sizes, int n_in,
                              void* d_out, int out_size, void* d_ws, size_t ws_size,
                              hipStream_t stream) {
    const float* x    = (const float*)d_in[0];
    const float* Wq   = (const float*)d_in[1];
    const float* bq   = (const float*)d_in[2];
    const float* Wk   = (const float*)d_in[3];
    const float* bk   = (const float*)d_in[4];
    const float* Wv   = (const float*)d_in[5];
    const float* bv   = (const float*)d_in[6];
    const float* Wo   = (const float*)d_in[7];
    const float* bo   = (const float*)d_in[8];
    const float* ln_g = (const float*)d_in[9];
    const float* ln_b = (const float*)d_in[10];
    float* out = (float*)d_out;   // sparse_embeddings has 0 elements; out = consistent_prompt

    // Workspace layout (floats): q, k, v each B*S*C; ksum S*C; y_pre B*S*C
    float* q    = (float*)d_ws;
    float* k    = q    + (size_t)BB * SS * CC;
    float* v    = k    + (size_t)BB * SS * CC;
    float* ksum = v    + (size_t)BB * SS * CC;
    float* ypre = ksum + (size_t)SS * CC;

    qkv_gemm_kernel<<<2048, 256, 0, stream>>>(x, Wq, bq, Wk, bk, Wv, bv, q, k, v);
    ksum_kernel<<<(SS * CC) / 256, 256, 0, stream>>>(k, ksum);
    attn_kernel<<<(BB * SS) / 8, 256, 0, stream>>>(q, ksum, v, ypre);
    proj_ln_kernel<<<2048, 256, 0, stream>>>(ypre, Wo, bo, ln_g, ln_b, x, out);
}